// MMAttention_36601711296727
// MI455X (gfx1250) — compile-verified
//
#include <hip/hip_runtime.h>

typedef __attribute__((ext_vector_type(2))) float v2f;
typedef __attribute__((ext_vector_type(8))) float v8f;

#define NB      2
#define NSEQ    4380
#define NPATH   284
#define NHIST   4096
#define DIMIN   512
#define HEADS   8
#define DHEAD   256
#define NOUT    6144          // 3 * HEADS * DHEAD
#define MROWS   8760          // NB * NSEQ
#define INNER   2048          // HEADS * DHEAD
#define SCALE_Q 0.0625f       // DHEAD^-0.5

#define WMMA_F32(acc, a, b) \
    acc = __builtin_amdgcn_wmma_f32_16x16x4_f32(false, a, false, b, (short)0, acc, false, false)

// ---------------------------------------------------------------------------
// K0: transpose w_qkv (512 x 6144) -> wT (6144 x 512) so the QKV GEMM can
// load B fragments as contiguous float2 (b64) instead of stride-6144 b32 pairs.
// ---------------------------------------------------------------------------
__global__ __launch_bounds__(256)
void transpose_w_kernel(const float* __restrict__ w, float* __restrict__ wT)
{
    int idx = blockIdx.x * 256 + threadIdx.x;      // over 512*6144
    if (idx >= DIMIN * NOUT) return;
    int c  = idx % NOUT;
    int kk = idx / NOUT;                           // coalesced read along c
    wT[(size_t)c * DIMIN + kk] = w[idx];
}

// ---------------------------------------------------------------------------
// K1: qkv = x @ w_qkv  (8760x512 @ 512x6144), scatter to head-major q,k,v.
// One wave computes a 32x64 tile: 2 A-fragments x 4 B-tiles = 8 accumulators.
// Per k-step: 2 A b64 + 4 B b64 loads : 8 WMMAs.  q pre-scaled by 1/16.
// ---------------------------------------------------------------------------
__global__ __launch_bounds__(32)
void qkv_gemm_kernel(const float* __restrict__ x, const float* __restrict__ wT,
                     float* __restrict__ q, float* __restrict__ k,
                     float* __restrict__ v)
{
    const int lane  = threadIdx.x & 31;
    const int rbase = blockIdx.y << 5;      // 32 rows / tile
    const int cbase = blockIdx.x << 6;      // 64 cols / tile
    const int mloc  = lane & 15;
    const int khalf = (lane >> 4) << 1;     // 0 or 2

    int ar0 = rbase + mloc;      if (ar0 >= MROWS) ar0 = MROWS - 1;
    int ar1 = rbase + 16 + mloc; if (ar1 >= MROWS) ar1 = MROWS - 1;
    const float* ap0 = x + (size_t)ar0 * DIMIN + khalf;
    const float* ap1 = x + (size_t)ar1 * DIMIN + khalf;

    const int c0 = cbase + mloc;            // < 6144 always
    const float* bp0 = wT + (size_t)(c0)      * DIMIN + khalf;
    const float* bp1 = wT + (size_t)(c0 + 16) * DIMIN + khalf;
    const float* bp2 = wT + (size_t)(c0 + 32) * DIMIN + khalf;
    const float* bp3 = wT + (size_t)(c0 + 48) * DIMIN + khalf;

    v8f a00 = {}, a01 = {}, a02 = {}, a03 = {};
    v8f a10 = {}, a11 = {}, a12 = {}, a13 = {};
    for (int k0 = 0; k0 < DIMIN; k0 += 4) {
        v2f f0 = *(const v2f*)(ap0 + k0);
        v2f f1 = *(const v2f*)(ap1 + k0);
        v2f b0 = *(const v2f*)(bp0 + k0);
        v2f b1 = *(const v2f*)(bp1 + k0);
        v2f b2 = *(const v2f*)(bp2 + k0);
        v2f b3 = *(const v2f*)(bp3 + k0);
        WMMA_F32(a00, f0, b0); WMMA_F32(a01, f0, b1);
        WMMA_F32(a02, f0, b2); WMMA_F32(a03, f0, b3);
        WMMA_F32(a10, f1, b0); WMMA_F32(a11, f1, b1);
        WMMA_F32(a12, f1, b2); WMMA_F32(a13, f1, b3);
    }

    const int rhalf = (lane >> 4) << 3;     // lanes 16-31 hold rows +8
    v8f accs[2][4] = { { a00, a01, a02, a03 }, { a10, a11, a12, a13 } };
    #pragma unroll
    for (int g = 0; g < 2; ++g) {
        #pragma unroll
        for (int t = 0; t < 4; ++t) {
            int c     = cbase + (t << 4) + mloc;
            int which = c >> 11;            // 0=q 1=k 2=v
            int inner = c & 2047;
            int h     = inner >> 8;
            int d     = inner & 255;
            float sc  = (which == 0) ? SCALE_Q : 1.0f;
            float* dst = (which == 0) ? q : ((which == 1) ? k : v);
            #pragma unroll
            for (int r = 0; r < 8; ++r) {
                int row = rbase + (g << 4) + rhalf + r;
                if (row < MROWS) {
                    int b = row / NSEQ;
                    int n = row - b * NSEQ;
                    dst[(((size_t)b * HEADS + h) * NSEQ + n) * DHEAD + d] =
                        accs[g][t][r] * sc;
                }
            }
        }
    }
}

// ---------------------------------------------------------------------------
// K2: C[z, i, j] = sum_d A[z, aOff+i, d] * B[z, bOff+j, d]   (NT GEMM, K=256)
// One wave computes a 32x64 tile; all fragments are contiguous b64 loads.
// ---------------------------------------------------------------------------
__global__ __launch_bounds__(32)
void attn_logits_kernel(const float* __restrict__ qbase,
                        const float* __restrict__ kbase,
                        float* __restrict__ cbaseptr,
                        int aOff, int bOff, int M, int Nn,
                        long long cStride, int ldc)
{
    const int z     = blockIdx.z;
    const int lane  = threadIdx.x & 31;
    const int rbase = blockIdx.y << 5;
    const int cb    = blockIdx.x << 6;
    const int mloc  = lane & 15;
    const int khalf = (lane >> 4) << 1;

    const float* A  = qbase + ((size_t)z * NSEQ + aOff) * DHEAD;
    const float* Bm = kbase + ((size_t)z * NSEQ + bOff) * DHEAD;
    float* C        = cbaseptr + (size_t)z * cStride;

    int ar0 = rbase + mloc;      if (ar0 >= M) ar0 = M - 1;
    int ar1 = rbase + 16 + mloc; if (ar1 >= M) ar1 = M - 1;
    const float* ap0 = A + (size_t)ar0 * DHEAD + khalf;
    const float* ap1 = A + (size_t)ar1 * DHEAD + khalf;

    int j0 = cb +      mloc; if (j0 >= Nn) j0 = Nn - 1;
    int j1 = cb + 16 + mloc; if (j1 >= Nn) j1 = Nn - 1;
    int j2 = cb + 32 + mloc; if (j2 >= Nn) j2 = Nn - 1;
    int j3 = cb + 48 + mloc; if (j3 >= Nn) j3 = Nn - 1;
    const float* bp0 = Bm + (size_t)j0 * DHEAD + khalf;
    const float* bp1 = Bm + (size_t)j1 * DHEAD + khalf;
    const float* bp2 = Bm + (size_t)j2 * DHEAD + khalf;
    const float* bp3 = Bm + (size_t)j3 * DHEAD + khalf;

    v8f a00 = {}, a01 = {}, a02 = {}, a03 = {};
    v8f a10 = {}, a11 = {}, a12 = {}, a13 = {};
    for (int k0 = 0; k0 < DHEAD; k0 += 4) {
        v2f f0 = *(const v2f*)(ap0 + k0);
        v2f f1 = *(const v2f*)(ap1 + k0);
        v2f b0 = *(const v2f*)(bp0 + k0);
        v2f b1 = *(const v2f*)(bp1 + k0);
        v2f b2 = *(const v2f*)(bp2 + k0);
        v2f b3 = *(const v2f*)(bp3 + k0);
        WMMA_F32(a00, f0, b0); WMMA_F32(a01, f0, b1);
        WMMA_F32(a02, f0, b2); WMMA_F32(a03, f0, b3);
        WMMA_F32(a10, f1, b0); WMMA_F32(a11, f1, b1);
        WMMA_F32(a12, f1, b2); WMMA_F32(a13, f1, b3);
    }

    const int rhalf = (lane >> 4) << 3;
    v8f accs[2][4] = { { a00, a01, a02, a03 }, { a10, a11, a12, a13 } };
    #pragma unroll
    for (int g = 0; g < 2; ++g) {
        #pragma unroll
        for (int t = 0; t < 4; ++t) {
            int col = cb + (t << 4) + mloc;
            if (col < Nn) {
                #pragma unroll
                for (int r = 0; r < 8; ++r) {
                    int row = rbase + (g << 4) + rhalf + r;
                    if (row < M) C[(size_t)row * ldc + col] = accs[g][t][r];
                }
            }
        }
    }
}

// ---------------------------------------------------------------------------
// K3: per-row softmax statistics (max, sum of exp) for fused softmax in PV.
// ---------------------------------------------------------------------------
__global__ __launch_bounds__(256)
void row_stats_kernel(const float* __restrict__ in, float* __restrict__ rmax,
                      float* __restrict__ rsum, int rowlen)
{
    __shared__ float red[256];
    const size_t row = blockIdx.x;
    const float* src = in + row * (size_t)rowlen;
    const int tid = threadIdx.x;

    float m = -3.0e38f;
    for (int i = tid; i < rowlen; i += 256) m = fmaxf(m, src[i]);
    red[tid] = m; __syncthreads();
    for (int s = 128; s > 0; s >>= 1) {
        if (tid < s) red[tid] = fmaxf(red[tid], red[tid + s]);
        __syncthreads();
    }
    m = red[0]; __syncthreads();

    float sum = 0.0f;
    for (int i = tid; i < rowlen; i += 256) sum += __expf(src[i] - m);
    red[tid] = sum; __syncthreads();
    for (int s = 128; s > 0; s >>= 1) {
        if (tid < s) red[tid] += red[tid + s];
        __syncthreads();
    }
    if (tid == 0) { rmax[row] = m; rsum[row] = red[0]; }
}

// ---------------------------------------------------------------------------
// K4: out_rows = softmax(P) @ V  with softmax applied on-the-fly to A
// fragments using precomputed row stats. Writes final (B, N, H*D) layout.
// 32x64 tile per wave: 2 A b64 + 8 B b32 loads : 8 WMMAs.
// ---------------------------------------------------------------------------
__global__ __launch_bounds__(32)
void attn_pv_kernel(const float* __restrict__ P, const float* __restrict__ vbase,
                    const float* __restrict__ rmax, const float* __restrict__ rsum,
                    float* __restrict__ out, int M, int K, int vOff, int n0)
{
    const int z = blockIdx.z, b = z >> 3, h = z & 7;
    const int lane  = threadIdx.x & 31;
    const int rbase = blockIdx.y << 5;
    const int cb    = blockIdx.x << 6;     // Nn = 256, always in range
    const int mloc  = lane & 15;
    const int khalf = (lane >> 4) << 1;

    const float* p = P + (size_t)z * M * K;
    const float* V = vbase + ((size_t)z * NSEQ + vOff) * DHEAD;
    float* o = out + ((size_t)b * NSEQ + n0) * INNER + h * DHEAD;

    int ar0 = rbase + mloc;      if (ar0 >= M) ar0 = M - 1;
    int ar1 = rbase + 16 + mloc; if (ar1 >= M) ar1 = M - 1;
    const float m0 = rmax[(size_t)z * M + ar0];
    const float i0 = 1.0f / rsum[(size_t)z * M + ar0];
    const float m1 = rmax[(size_t)z * M + ar1];
    const float i1 = 1.0f / rsum[(size_t)z * M + ar1];
    const float* ap0 = p + (size_t)ar0 * K + khalf;
    const float* ap1 = p + (size_t)ar1 * K + khalf;
    const float* bpc = V + cb + mloc;

    v8f a00 = {}, a01 = {}, a02 = {}, a03 = {};
    v8f a10 = {}, a11 = {}, a12 = {}, a13 = {};
    for (int k0 = 0; k0 < K; k0 += 4) {
        v2f f0 = *(const v2f*)(ap0 + k0);
        v2f f1 = *(const v2f*)(ap1 + k0);
        f0.x = __expf(f0.x - m0) * i0;
        f0.y = __expf(f0.y - m0) * i0;
        f1.x = __expf(f1.x - m1) * i1;
        f1.y = __expf(f1.y - m1) * i1;
        const float* bp = bpc + (size_t)(k0 + khalf) * DHEAD;
        v2f b0, b1, b2, b3;
        b0.x = bp[0];   b0.y = bp[DHEAD];
        b1.x = bp[16];  b1.y = bp[DHEAD + 16];
        b2.x = bp[32];  b2.y = bp[DHEAD + 32];
        b3.x = bp[48];  b3.y = bp[DHEAD + 48];
        WMMA_F32(a00, f0, b0); WMMA_F32(a01, f0, b1);
        WMMA_F32(a02, f0, b2); WMMA_F32(a03, f0, b3);
        WMMA_F32(a10, f1, b0); WMMA_F32(a11, f1, b1);
        WMMA_F32(a12, f1, b2); WMMA_F32(a13, f1, b3);
    }

    const int rhalf = (lane >> 4) << 3;
    v8f accs[2][4] = { { a00, a01, a02, a03 }, { a10, a11, a12, a13 } };
    #pragma unroll
    for (int g = 0; g < 2; ++g) {
        #pragma unroll
        for (int t = 0; t < 4; ++t) {
            int col = cb + (t << 4) + mloc;
            #pragma unroll
            for (int r = 0; r < 8; ++r) {
                int row = rbase + (g << 4) + rhalf + r;
                if (row < M) o[(size_t)row * INNER + col] = accs[g][t][r];
            }
        }
    }
}

// ---------------------------------------------------------------------------
// K5: depthwise conv residual along the sequence dim, accumulated into out.
// ---------------------------------------------------------------------------
__global__ __launch_bounds__(256)
void conv_res_kernel(const float* __restrict__ v, const float* __restrict__ wconv,
                     float* __restrict__ out)
{
    size_t idx = (size_t)blockIdx.x * 256 + threadIdx.x;
    if (idx >= (size_t)NB * HEADS * NSEQ * DHEAD) return;
    int d = idx & 255;
    size_t t = idx >> 8;
    int n = (int)(t % NSEQ); t /= NSEQ;
    int h = (int)(t & 7);
    int b = (int)(t >> 3);

    const float* vp = v + ((size_t)b * HEADS + h) * NSEQ * DHEAD + d;
    float acc = 0.0f;
    #pragma unroll
    for (int kt = 0; kt < 33; ++kt) {
        int nn = n + kt - 16;
        if (nn >= 0 && nn < NSEQ) acc += vp[(size_t)nn * DHEAD] * wconv[h * 33 + kt];
    }
    out[((size_t)b * NSEQ + n) * INNER + h * DHEAD + d] += acc;
}

// ---------------------------------------------------------------------------
extern "C" void kernel_launch(void* const* d_in, const int* in_sizes, int n_in,
                              void* d_out, int out_size, void* d_ws, size_t ws_size,
                              hipStream_t stream)
{
    const float* x      = (const float*)d_in[0];
    const float* w_qkv  = (const float*)d_in[1];
    const float* w_conv = (const float*)d_in[2];
    float* out = (float*)d_out;
    float* ws  = (float*)d_ws;

    const size_t QKV   = (size_t)NB * HEADS * NSEQ * DHEAD; // 17,940,480
    const size_t AP_SZ = (size_t)16 * NPATH * NPATH;        // 1,290,496
    const size_t CP_SZ = (size_t)16 * NPATH * NHIST;        // 18,612,224

    // workspace: head-major q, k, v + softmax row stats + transposed weights
    float* q     = ws;
    float* k     = ws + QKV;
    float* v     = ws + 2 * QKV;
    float* cpMax = ws + 3 * QKV;
    float* cpSum = cpMax + (size_t)16 * NPATH;
    float* chMax = cpSum + (size_t)16 * NPATH;
    float* chSum = chMax + (size_t)16 * NHIST;
    float* wT    = chSum + (size_t)16 * NHIST;              // 6144*512 floats

    // d_out layout: [out | attn_pathways | cross_attn_pathways | pre_softmax_hist]
    float* ap_out = out + QKV;
    float* cp_out = ap_out + AP_SZ;
    float* ch_out = cp_out + CP_SZ;

    // K0: transpose weights (3,145,728 elements)
    transpose_w_kernel<<<dim3((DIMIN * NOUT + 255) / 256, 1, 1), 256, 0, stream>>>(
        w_qkv, wT);

    // K1: QKV GEMM (6144/64=96 col tiles, ceil(8760/32)=274 row tiles)
    qkv_gemm_kernel<<<dim3(96, 274, 1), 32, 0, stream>>>(x, wT, q, k, v);

    // K2: attention logits (all three straight into d_out)
    // attn_pathways: q_p . k_p^T  (284 x 284)
    attn_logits_kernel<<<dim3(5, 9, 16), 32, 0, stream>>>(
        q, k, ap_out, 0, 0, NPATH, NPATH, (long long)NPATH * NPATH, NPATH);
    // cross_attn_pathways: q_p . k_h^T  (284 x 4096)
    attn_logits_kernel<<<dim3(64, 9, 16), 32, 0, stream>>>(
        q, k, cp_out, 0, NPATH, NPATH, NHIST, (long long)NPATH * NHIST, NHIST);
    // cross_attn_histology: q_h . k_p^T  (4096 x 284)
    attn_logits_kernel<<<dim3(5, 128, 16), 32, 0, stream>>>(
        q, k, ch_out, NPATH, 0, NHIST, NPATH, (long long)NHIST * NPATH, NPATH);

    // K3: softmax row stats
    row_stats_kernel<<<dim3(16 * NPATH, 1, 1), 256, 0, stream>>>(cp_out, cpMax, cpSum, NHIST);
    row_stats_kernel<<<dim3(16 * NHIST, 1, 1), 256, 0, stream>>>(ch_out, chMax, chSum, NPATH);

    // K4: out_pathways = softmax(cross_attn_pathways) @ v_h  -> rows [0,284)
    attn_pv_kernel<<<dim3(4, 9, 16), 32, 0, stream>>>(
        cp_out, v, cpMax, cpSum, out, NPATH, NHIST, NPATH, 0);
    //     out_histology = softmax(cross_attn_histology) @ v_p -> rows [284,4380)
    attn_pv_kernel<<<dim3(4, 128, 16), 32, 0, stream>>>(
        ch_out, v, chMax, chSum, out, NHIST, NPATH, 0, NPATH);

    // K5: depthwise conv residual added into out
    const size_t total = (size_t)NB * HEADS * NSEQ * DHEAD;
    conv_res_kernel<<<dim3((unsigned)((total + 255) / 256), 1, 1), 256, 0, stream>>>(
        v, w_conv, out);
}